// TensorMemory_37211596653036
// MI455X (gfx1250) — compile-verified
//
#include <hip/hip_runtime.h>

typedef __attribute__((ext_vector_type(2))) float v2f;
typedef __attribute__((ext_vector_type(4))) float v4f;
typedef __attribute__((ext_vector_type(8))) float v8f;

#define S_LEN 4096
#define D_LEN 128
#define N_HEADS 32          // B*H = 2*16
#define EPS_CONST 1e-6f
#define LDSS 132            // padded LDS row stride (floats); 132*4=528B, 16B aligned
#define CH 32               // S-chunk rows per iteration (phase 1)

// Issue one 128-bit global->LDS async copy (GVS form, ASYNCcnt-tracked).
// ldsoff: byte offset of destination within wave's LDS (low 32 bits of the
// generic pointer to a __shared__ object, per the flat->LDS aperture rule).
// goff:   per-lane byte offset from the uniform 64-bit base in SGPRs.
__device__ __forceinline__ void async_copy_b128(unsigned ldsoff, unsigned goff,
                                                const void* base_uniform) {
    asm volatile("global_load_async_to_lds_b128 %0, %1, %2"
                 :: "v"(ldsoff), "v"(goff), "s"(base_uniform) : "memory");
}
__device__ __forceinline__ void wait_asynccnt_le8() {
    asm volatile("s_wait_asynccnt 0x8" ::: "memory");
}
__device__ __forceinline__ void wait_asynccnt_0() {
    asm volatile("s_wait_asynccnt 0x0" ::: "memory");
}

// ---------------------------------------------------------------------------
// Phase 1: per head  M = M0 + V^T K   (128x128),  z = z0 + sum_s K[s]
// 32 blocks (one per head), 256 threads = 8 waves in a 4x2 grid: each wave
// owns a 32(row d) x 64(col e) super-tile = 2x4 WMMA tiles.  K/V stream
// through double-buffered LDS via async global->LDS copies.
// ---------------------------------------------------------------------------
__global__ __launch_bounds__(256) void tm_update_kernel(
    const float* __restrict__ K, const float* __restrict__ V,
    const float* __restrict__ M0, const float* __restrict__ z0,
    float* __restrict__ M, float* __restrict__ z)
{
    __shared__ __align__(16) float sK[2][CH][LDSS];
    __shared__ __align__(16) float sV[2][CH][LDSS];
    __shared__ float sZ[2][D_LEN];

    const int head = blockIdx.x;
    const int tid  = threadIdx.x;
    const int wave = tid >> 5;
    const int lane = tid & 31;
    const int n16  = lane & 15;     // N/M index within a 16x16 tile
    const int kh   = lane >> 4;     // lane-half: K pair (A/B) / row half (C/D)
    const int wr   = wave >> 1;     // 0..3 : row super-tile
    const int wc   = wave & 1;      // 0..1 : col super-tile

    const float* __restrict__ Kh = K + (size_t)head * S_LEN * D_LEN;
    const float* __restrict__ Vh = V + (size_t)head * S_LEN * D_LEN;

    // staging coords: thread covers rows {r0, r0+8, r0+16, r0+24} at col c0
    const int r0 = (tid * 4) >> 7;
    const int c0 = (tid * 4) & 127;

    v8f acc[2][4] = {};
    float zacc = 0.0f;
    const int zcol = tid & 127;
    const int zrh  = tid >> 7;

    auto issue_chunk = [&](int s0, int buf) {
        #pragma unroll
        for (int i = 0; i < 4; ++i) {
            const int row = r0 + 8 * i;
            const unsigned goff =
                (unsigned)(((s0 + row) * D_LEN + c0) * sizeof(float));
            async_copy_b128((unsigned)(uintptr_t)&sK[buf][row][c0], goff, Kh);
            async_copy_b128((unsigned)(uintptr_t)&sV[buf][row][c0], goff, Vh);
        }
    };

    issue_chunk(0, 0);                           // prologue

    for (int s0 = 0; s0 < S_LEN; s0 += CH) {
        const int buf = (s0 >> 5) & 1;
        if (s0 + CH < S_LEN) {
            issue_chunk(s0 + CH, buf ^ 1);       // prefetch next into other buffer
            wait_asynccnt_le8();                 // current chunk's 8 copies done
        } else {
            wait_asynccnt_0();
        }
        __syncthreads();

        // z partial sums over this K chunk
        #pragma unroll
        for (int r = 0; r < 16; ++r) zacc += sK[buf][zrh * 16 + r][zcol];

        // WMMA: 8 K=4 steps; 2 A operands, 4 B operands, 8 MACs per step
        #pragma unroll
        for (int ks = 0; ks < CH; ks += 4) {
            v2f a[2], b[4];
            #pragma unroll
            for (int r = 0; r < 2; ++r) {        // A[m=d, k=s] = V[s, d]
                a[r].x = sV[buf][ks + 2 * kh + 0][wr * 32 + r * 16 + n16];
                a[r].y = sV[buf][ks + 2 * kh + 1][wr * 32 + r * 16 + n16];
            }
            #pragma unroll
            for (int c = 0; c < 4; ++c) {        // B[k=s, n=e] = K[s, e]
                b[c].x = sK[buf][ks + 2 * kh + 0][wc * 64 + c * 16 + n16];
                b[c].y = sK[buf][ks + 2 * kh + 1][wc * 64 + c * 16 + n16];
            }
            #pragma unroll
            for (int r = 0; r < 2; ++r)
                #pragma unroll
                for (int c = 0; c < 4; ++c)
                    acc[r][c] = __builtin_amdgcn_wmma_f32_16x16x4_f32(
                        false, a[r], false, b[c], (short)0, acc[r][c],
                        false, false);
        }
        __syncthreads();
    }

    // M = M0 + acc
    const float* __restrict__ M0h = M0 + (size_t)head * D_LEN * D_LEN;
    float* __restrict__ Mh = M + (size_t)head * D_LEN * D_LEN;
    #pragma unroll
    for (int r = 0; r < 2; ++r)
        #pragma unroll
        for (int c = 0; c < 4; ++c)
            #pragma unroll
            for (int j = 0; j < 8; ++j) {
                const int d = wr * 32 + r * 16 + kh * 8 + j;
                const int e = wc * 64 + c * 16 + n16;
                Mh[(size_t)d * D_LEN + e] =
                    M0h[(size_t)d * D_LEN + e] + acc[r][c][j];
            }

    // z = z0 + reduced partials
    sZ[zrh][zcol] = zacc;
    __syncthreads();
    if (tid < D_LEN)
        z[head * D_LEN + tid] = z0[head * D_LEN + tid] + sZ[0][tid] + sZ[1][tid];
}

// ---------------------------------------------------------------------------
// Phase 2: out[s,d] = (sum_e M[d,e] Q[s,e]) / (sum_d z[d] Q[s,d] + eps)
// grid = (S/128, 32 heads); block does a 128(s) x 128(d) tile; waves in the
// same 4x2 arrangement (32 s-rows x 64 d-cols each).
// ---------------------------------------------------------------------------
__global__ __launch_bounds__(256) void tm_retrieve_kernel(
    const float* __restrict__ Q, const float* __restrict__ M,
    const float* __restrict__ z, float* __restrict__ out)
{
    __shared__ __align__(16) float sQ [D_LEN][LDSS];
    __shared__ __align__(16) float sMT[D_LEN][LDSS];   // sMT[e][d] = M[d][e]
    __shared__ float sZ[D_LEN];
    __shared__ float sDen[D_LEN];

    const int head = blockIdx.y;
    const int s0   = blockIdx.x * 128;
    const int tid  = threadIdx.x;
    const int wave = tid >> 5;
    const int lane = tid & 31;
    const int n16  = lane & 15;
    const int kh   = lane >> 4;
    const int wr   = wave >> 1;
    const int wc   = wave & 1;

    const float* __restrict__ Qh = Q + (size_t)head * S_LEN * D_LEN
                                     + (size_t)s0 * D_LEN;
    const float* __restrict__ Mh = M + (size_t)head * D_LEN * D_LEN;

    // Q tile: async global->LDS (128x128, 16 B128 copies per thread)
    const int r0 = (tid * 4) >> 7;
    const int c0 = (tid * 4) & 127;
    #pragma unroll
    for (int i = 0; i < 16; ++i) {
        const int row = r0 + 8 * i;
        const unsigned goff = (unsigned)((row * D_LEN + c0) * sizeof(float));
        async_copy_b128((unsigned)(uintptr_t)&sQ[row][c0], goff, Qh);
    }

    // M: VGPR-mediated transpose into LDS (no 32-bit TR load exists)
    #pragma unroll
    for (int i = 0; i < 16; ++i) {
        const int linear = (tid + i * 256) * 4;
        const int d = linear >> 7;
        const int e = linear & 127;
        const v4f m4 = *(const v4f*)(Mh + linear);
        sMT[e + 0][d] = m4.x;
        sMT[e + 1][d] = m4.y;
        sMT[e + 2][d] = m4.z;
        sMT[e + 3][d] = m4.w;
    }
    if (tid < D_LEN) sZ[tid] = z[head * D_LEN + tid];
    wait_asynccnt_0();
    __syncthreads();

    // denominators: den[s] = eps + sum_d Q[s,d]*z[d]
    if (tid < D_LEN) {
        float den = EPS_CONST;
        #pragma unroll 8
        for (int d = 0; d < D_LEN; ++d) den += sQ[tid][d] * sZ[d];
        sDen[tid] = den;
    }

    // numerator GEMM: 32 K=4 steps, 2x4 tiles per wave
    v8f acc[2][4] = {};
    #pragma unroll
    for (int kb = 0; kb < D_LEN; kb += 4) {
        v2f a[2], b[4];
        #pragma unroll
        for (int r = 0; r < 2; ++r) {            // A[m=s, k=e] = Q[s, e]
            a[r].x = sQ[wr * 32 + r * 16 + n16][kb + 2 * kh + 0];
            a[r].y = sQ[wr * 32 + r * 16 + n16][kb + 2 * kh + 1];
        }
        #pragma unroll
        for (int c = 0; c < 4; ++c) {            // B[k=e, n=d] = M[d, e]
            b[c].x = sMT[kb + 2 * kh + 0][wc * 64 + c * 16 + n16];
            b[c].y = sMT[kb + 2 * kh + 1][wc * 64 + c * 16 + n16];
        }
        #pragma unroll
        for (int r = 0; r < 2; ++r)
            #pragma unroll
            for (int c = 0; c < 4; ++c)
                acc[r][c] = __builtin_amdgcn_wmma_f32_16x16x4_f32(
                    false, a[r], false, b[c], (short)0, acc[r][c],
                    false, false);
    }
    __syncthreads();   // sDen complete before use

    // divide and store
    float* __restrict__ Oh = out + (size_t)head * S_LEN * D_LEN
                                 + (size_t)s0 * D_LEN;
    #pragma unroll
    for (int r = 0; r < 2; ++r)
        #pragma unroll
        for (int c = 0; c < 4; ++c)
            #pragma unroll
            for (int j = 0; j < 8; ++j) {
                const int srow = wr * 32 + r * 16 + kh * 8 + j;
                const int dcol = wc * 64 + c * 16 + n16;
                Oh[(size_t)srow * D_LEN + dcol] = acc[r][c][j] / sDen[srow];
            }
}

// ---------------------------------------------------------------------------
extern "C" void kernel_launch(void* const* d_in, const int* in_sizes, int n_in,
                              void* d_out, int out_size, void* d_ws, size_t ws_size,
                              hipStream_t stream) {
    const float* keys    = (const float*)d_in[0];
    const float* values  = (const float*)d_in[1];
    const float* queries = (const float*)d_in[2];
    const float* M0      = (const float*)d_in[3];
    const float* z0      = (const float*)d_in[4];
    float* outp = (float*)d_out;

    float* Mw = (float*)d_ws;                              // 32*128*128 floats = 2 MB
    float* zw = Mw + (size_t)N_HEADS * D_LEN * D_LEN;      // 32*128 floats

    tm_update_kernel<<<dim3(N_HEADS), dim3(256), 0, stream>>>(
        keys, values, M0, z0, Mw, zw);
    tm_retrieve_kernel<<<dim3(S_LEN / 128, N_HEADS), dim3(256), 0, stream>>>(
        queries, Mw, zw, outp);
}